// BaseTransform_9294309228775
// MI455X (gfx1250) — compile-verified
//
#include <hip/hip_runtime.h>

// Problem constants (fixed by the reference)
#define NCAM 6
#define ND_  118
#define FH_  32
#define FW_  88
#define C_   80
#define NX0  360
#define NX1  360
#define NZ_  1
#define PTS_PER_CAM (ND_ * FH_ * FW_)   // 332288
#define NP (NCAM * PTS_PER_CAM)         // 1993728  (divisible by 16; per-cam count divisible by 16)
#define OUT_ELEMS (C_ * NX0 * NX1)      // 10368000
#define VOX_OFFSET_BYTES 1024

typedef __attribute__((ext_vector_type(2))) float v2f;
typedef __attribute__((ext_vector_type(8))) float v8f;

__device__ inline void inv3(const float m[9], float o[9]) {
    float a = m[0], b = m[1], c = m[2];
    float d = m[3], e = m[4], f = m[5];
    float g = m[6], h = m[7], i = m[8];
    float A = e * i - f * h;
    float B = -(d * i - f * g);
    float Cc = d * h - e * g;
    float det = a * A + b * B + c * Cc;
    float r = 1.0f / det;
    o[0] = A * r;               o[1] = -(b * i - c * h) * r; o[2] = (b * f - c * e) * r;
    o[3] = B * r;               o[4] = (a * i - c * g) * r;  o[5] = -(a * f - c * d) * r;
    o[6] = Cc * r;              o[7] = -(a * h - b * g) * r; o[8] = (a * e - b * d) * r;
}

// Build per-camera folded affine maps:
//   A1 (3x4): p' = inv(post_rot) * ((u,v,d) - post_trans)
//   A2 (3x4): coord = diag(1/dx) * E * R * inv(K) * q + (E*tc + te - lo)/dx
__global__ void prep_kernel(const float* __restrict__ intr,
                            const float* __restrict__ c2l,
                            const float* __restrict__ ida,
                            const float* __restrict__ bda,
                            float* __restrict__ mats) {
    int n = threadIdx.x;
    if (n >= NCAM) return;
    const float* In = intr + n * 16;
    const float* Cn = c2l + n * 16;
    const float* Pn = ida + n * 16;
    const float* En = bda;  // B == 1
    float P[9], K[9], R[9], E[9], tp[3], tc[3], te[3];
    for (int r = 0; r < 3; r++)
        for (int c = 0; c < 3; c++) {
            P[r * 3 + c] = Pn[r * 4 + c];
            K[r * 3 + c] = In[r * 4 + c];
            R[r * 3 + c] = Cn[r * 4 + c];
            E[r * 3 + c] = En[r * 4 + c];
        }
    for (int r = 0; r < 3; r++) {
        tp[r] = Pn[r * 4 + 3];
        tc[r] = Cn[r * 4 + 3];
        te[r] = En[r * 4 + 3];
    }
    float Pi[9], Ki[9];
    inv3(P, Pi);
    inv3(K, Ki);
    float* A1 = mats + n * 24;
    float* A2 = mats + n * 24 + 12;
    for (int r = 0; r < 3; r++) {
        A1[r * 4 + 0] = Pi[r * 3 + 0];
        A1[r * 4 + 1] = Pi[r * 3 + 1];
        A1[r * 4 + 2] = Pi[r * 3 + 2];
        A1[r * 4 + 3] = -(Pi[r * 3 + 0] * tp[0] + Pi[r * 3 + 1] * tp[1] + Pi[r * 3 + 2] * tp[2]);
    }
    float G[9], M2[9];
    for (int r = 0; r < 3; r++)
        for (int c = 0; c < 3; c++)
            G[r * 3 + c] = E[r * 3 + 0] * R[0 * 3 + c] + E[r * 3 + 1] * R[1 * 3 + c] + E[r * 3 + 2] * R[2 * 3 + c];
    for (int r = 0; r < 3; r++)
        for (int c = 0; c < 3; c++)
            M2[r * 3 + c] = G[r * 3 + 0] * Ki[0 * 3 + c] + G[r * 3 + 1] * Ki[1 * 3 + c] + G[r * 3 + 2] * Ki[2 * 3 + c];
    const float dxv[3] = {0.3f, 0.3f, 20.0f};
    const float lov[3] = {-54.0f, -54.0f, -10.0f};
    for (int r = 0; r < 3; r++) {
        float tE = E[r * 3 + 0] * tc[0] + E[r * 3 + 1] * tc[1] + E[r * 3 + 2] * tc[2] + te[r];
        float inv = 1.0f / dxv[r];
        A2[r * 4 + 0] = M2[r * 3 + 0] * inv;
        A2[r * 4 + 1] = M2[r * 3 + 1] * inv;
        A2[r * 4 + 2] = M2[r * 3 + 2] * inv;
        A2[r * 4 + 3] = (tE - lov[r]) * inv;
    }
}

// Geometry via V_WMMA_F32_16X16X4_F32: one wave transforms 16 points per WMMA.
// A (16x4): rows 0-2 = A2 (voxel-space affine), rows 3-15 zero.
// B (4x16): columns = homogeneous q = (px*pz, py*pz, pz, 1) for 16 points.
// D (16x16): rows 0-2 = voxel coords; lane i (0-15) holds point i's coords in D[0..2].
__global__ void __launch_bounds__(256) geom_wmma_kernel(const float* __restrict__ mats,
                                                        int* __restrict__ vox) {
    int tid = blockIdx.x * blockDim.x + threadIdx.x;
    int wave = tid >> 5;
    int lane = tid & 31;
    int i = lane & 15;
    bool hi = lane >= 16;
    int p = wave * 16 + i;  // launch sized so p < NP always
    int n = p / PTS_PER_CAM;
    int rem = p - n * PTS_PER_CAM;
    int di = rem / (FH_ * FW_);
    int r2 = rem - di * (FH_ * FW_);
    int h = r2 / FW_;
    int w = r2 - h * FW_;
    float u = (float)w * (703.0f / 87.0f);   // linspace(0, IW-1, FW)
    float v = (float)h * (255.0f / 31.0f);   // linspace(0, IH-1, FH)
    float dep = 1.0f + 0.5f * (float)di;     // arange(1, 60, 0.5)
    const float* A1 = mats + n * 24;
    const float* A2 = mats + n * 24 + 12;
    float px = A1[0] * u + A1[1] * v + A1[2] * dep + A1[3];
    float py = A1[4] * u + A1[5] * v + A1[6] * dep + A1[7];
    float pz = A1[8] * u + A1[9] * v + A1[10] * dep + A1[11];
    float qx = px * pz, qy = py * pz, qz = pz;
    // B operand layout (32-bit, 4x16): VGPR0 = rows K=0 (lanes 0-15) / K=2 (lanes 16-31),
    //                                  VGPR1 = rows K=1 / K=3.
    v2f bop;
    bop.x = hi ? qz : qx;
    bop.y = hi ? 1.0f : qy;
    // A operand layout (32-bit, 16x4): VGPR0 = K=0 / K=2, VGPR1 = K=1 / K=3, M = lane&15.
    float rowon = (i < 3) ? 1.0f : 0.0f;
    int mr = (i < 3) ? i : 0;
    v2f aop;
    aop.x = (hi ? A2[mr * 4 + 2] : A2[mr * 4 + 0]) * rowon;
    aop.y = (hi ? A2[mr * 4 + 3] : A2[mr * 4 + 1]) * rowon;
    v8f acc = {0.f, 0.f, 0.f, 0.f, 0.f, 0.f, 0.f, 0.f};
    v8f dres = __builtin_amdgcn_wmma_f32_16x16x4_f32(
        /*neg_a=*/false, aop, /*neg_b=*/false, bop,
        /*c_mod=*/(short)0, acc, /*reuse_a=*/false, /*reuse_b=*/false);
    if (!hi) {
        float cx = dres[0], cy = dres[1], cz = dres[2];
        int ix = (int)cx;  // trunc-toward-zero matches astype(int32)
        int iy = (int)cy;
        int iz = (int)cz;
        bool kept = (ix >= 0) & (ix < NX0) & (iy >= 0) & (iy < NX1) & (iz >= 0) & (iz < NZ_);
        vox[p] = kept ? (ix * NX1 + iy) : -1;
    }
}

// Streaming scatter: 16 points x 80 channels per 256-thread block.
// Reads of x are fully contiguous (1280 floats per block); atomics land in L2
// (output region is 41.5 MB << 192 MB L2).
__global__ void __launch_bounds__(256) scatter_kernel(const float* __restrict__ x,
                                                      const int* __restrict__ vox,
                                                      float* __restrict__ out) {
    int p0 = blockIdx.x * 16;
    int tid = threadIdx.x;
#pragma unroll
    for (int it = 0; it < 5; ++it) {
        int e = it * 256 + tid;     // 0..1279
        int pl = e / C_;
        int c = e - pl * C_;
        int p = p0 + pl;
        int vv = vox[p];
        float val = x[(size_t)p * C_ + c];
        if (vv >= 0) atomicAdd(out + (size_t)c * (NX0 * NX1) + vv, val);
    }
}

// Fallback if the workspace cannot hold the vox array: recompute geometry inline.
__global__ void __launch_bounds__(256) fused_kernel(const float* __restrict__ x,
                                                    const float* __restrict__ mats,
                                                    float* __restrict__ out) {
    int p0 = blockIdx.x * 16;
    int tid = threadIdx.x;
#pragma unroll
    for (int it = 0; it < 5; ++it) {
        int e = it * 256 + tid;
        int pl = e / C_;
        int c = e - pl * C_;
        int p = p0 + pl;
        int n = p / PTS_PER_CAM;
        int rem = p - n * PTS_PER_CAM;
        int di = rem / (FH_ * FW_);
        int r2 = rem - di * (FH_ * FW_);
        int h = r2 / FW_;
        int w = r2 - h * FW_;
        float u = (float)w * (703.0f / 87.0f);
        float v = (float)h * (255.0f / 31.0f);
        float dep = 1.0f + 0.5f * (float)di;
        const float* A1 = mats + n * 24;
        const float* A2 = mats + n * 24 + 12;
        float px = A1[0] * u + A1[1] * v + A1[2] * dep + A1[3];
        float py = A1[4] * u + A1[5] * v + A1[6] * dep + A1[7];
        float pz = A1[8] * u + A1[9] * v + A1[10] * dep + A1[11];
        float qx = px * pz, qy = py * pz, qz = pz;
        float cx = A2[0] * qx + A2[1] * qy + A2[2] * qz + A2[3];
        float cy = A2[4] * qx + A2[5] * qy + A2[6] * qz + A2[7];
        float cz = A2[8] * qx + A2[9] * qy + A2[10] * qz + A2[11];
        int ix = (int)cx, iy = (int)cy, iz = (int)cz;
        bool kept = (ix >= 0) & (ix < NX0) & (iy >= 0) & (iy < NX1) & (iz >= 0) & (iz < NZ_);
        float val = x[(size_t)p * C_ + c];
        if (kept) atomicAdd(out + (size_t)c * (NX0 * NX1) + ix * NX1 + iy, val);
    }
}

__global__ void zero_kernel(float4* __restrict__ out, int n4) {
    int i = blockIdx.x * blockDim.x + threadIdx.x;
    if (i < n4) out[i] = make_float4(0.f, 0.f, 0.f, 0.f);
}

extern "C" void kernel_launch(void* const* d_in, const int* in_sizes, int n_in,
                              void* d_out, int out_size, void* d_ws, size_t ws_size,
                              hipStream_t stream) {
    const float* x    = (const float*)d_in[0];
    const float* intr = (const float*)d_in[1];
    const float* c2l  = (const float*)d_in[2];
    const float* ida  = (const float*)d_in[3];
    const float* bda  = (const float*)d_in[4];
    float* out = (float*)d_out;
    float* mats = (float*)d_ws;
    int* vox = (int*)((char*)d_ws + VOX_OFFSET_BYTES);

    // Scatter accumulates -> must zero the output every call (graph replays).
    int n4 = OUT_ELEMS / 4;
    zero_kernel<<<(n4 + 255) / 256, 256, 0, stream>>>((float4*)out, n4);
    prep_kernel<<<1, 8, 0, stream>>>(intr, c2l, ida, bda, mats);

    size_t need = (size_t)VOX_OFFSET_BYTES + (size_t)NP * sizeof(int);
    if (ws_size >= need) {
        // 16 points per wave, 8 waves per block: NP/16/8 = 15576 blocks.
        geom_wmma_kernel<<<NP / 16 / 8, 256, 0, stream>>>(mats, vox);
        scatter_kernel<<<NP / 16, 256, 0, stream>>>(x, vox, out);
    } else {
        fused_kernel<<<NP / 16, 256, 0, stream>>>(x, mats, out);
    }
}